// BaseNet_72533407694985
// MI455X (gfx1250) — compile-verified
//
#include <hip/hip_runtime.h>
#include <hip/hip_bf16.h>

// ---------------- problem constants ----------------
#define BATCH      32
#define NPTS       131072
#define ROWS_PB    2048                    // rows per block tile
#define BLKS_PB    (NPTS / ROWS_PB)        // 64 blocks per batch
#define TILE_F     (ROWS_PB * 6)           // 12288 floats = 48KB LDS tile
// workspace layout (uint32): [0..2047] counts, [2048..4095] offsets, [4096..4127] totals
#define WS_OFFSETS 2048u
#define WS_TOTALS  4096u

typedef float v2f  __attribute__((ext_vector_type(2)));
typedef float v8f  __attribute__((ext_vector_type(8)));
typedef unsigned int u32x4 __attribute__((ext_vector_type(4)));
typedef int i32x4 __attribute__((ext_vector_type(4)));
typedef int i32x8 __attribute__((ext_vector_type(8)));

#if defined(__gfx1250__) && __has_builtin(__builtin_amdgcn_tensor_load_to_lds)
#define HAVE_TDM 1
#else
#define HAVE_TDM 0
#endif

#if defined(__gfx1250__) && __has_builtin(__builtin_amdgcn_wmma_f32_16x16x4_f32)
#define HAVE_WMMA 1
#else
#define HAVE_WMMA 0
#endif

__device__ __forceinline__ unsigned ballot32(bool p) {
#if defined(__gfx1250__) && __has_builtin(__builtin_amdgcn_ballot_w32)
  return __builtin_amdgcn_ballot_w32(p);
#else
  return p ? 1u : 0u;  // host-pass placeholder only
#endif
}

// ---- Stage one contiguous 48KB tile (2048 rows x 6 f32) global -> LDS ----
// TDM: 1D tile of 12288 dwords. D# per CDNA5 ISA ch.8 bit layout.
__device__ __forceinline__ void stage_tile(const float* __restrict__ gsrc, float* tile) {
#if HAVE_TDM
  if (threadIdx.x < 32) {  // wave 0 issues the DMA (EXEC ignored by TDM)
    unsigned lds_addr = (unsigned)(unsigned long long)(void*)tile;
    unsigned long long ga = (unsigned long long)(const void*)gsrc;
    u32x4 g0;
    g0[0] = 1u;                                  // count = 1 valid descriptor
    g0[1] = lds_addr;                            // lds_addr [63:32]
    g0[2] = (unsigned)(ga & 0xffffffffull);      // global_addr [95:64]
    g0[3] = (unsigned)((ga >> 32) & 0x01ffffffull) | (2u << 30); // addr[56:32] | type=2
    i32x8 g1;
    g1[0] = (int)(2u << 16);                     // data_size = 4B
    g1[1] = (int)((unsigned)TILE_F << 16);       // tensor_dim0 low16 at [63:48]
    g1[2] = (int)(1u << 16);                     // tensor_dim0 hi = 0, tensor_dim1 = 1
    g1[3] = (int)((unsigned)TILE_F << 16);       // tile_dim0 = 12288 at [127:112]
    g1[4] = 1;                                   // tile_dim1 = 1, tile_dim2 = 0
    g1[5] = (int)TILE_F;                         // tensor_dim0_stride low32
    g1[6] = 0;
    g1[7] = 0;
    i32x4 z4 = {0, 0, 0, 0};
    i32x8 z8 = {0, 0, 0, 0, 0, 0, 0, 0};
    // clang-23 / therock-10.0 6-arg form: (g0, g1, g2, g3, extra, cpol)
    __builtin_amdgcn_tensor_load_to_lds(g0, g1, z4, z4, z8, 0);
    __builtin_amdgcn_s_wait_tensorcnt(0);
  }
#else
  const float4* s = (const float4*)gsrc;         // 49152B tile is 16B-aligned
  float4* d = (float4*)tile;
  for (unsigned i = threadIdx.x; i < TILE_F / 4; i += blockDim.x) d[i] = s[i];
#endif
  __syncthreads();
}

// A-matrix (16x4 f32) = [ R^T | t ] so that D = A x B gives D[e,n] = p[n,e].
// 32-bit A layout: lanes 0-15 hold M, VGPR0 = {K0 | K2}, VGPR1 = {K1 | K3}.
__device__ __forceinline__ v2f build_A(const float* __restrict__ T, unsigned lane) {
  v2f a; a.x = 0.0f; a.y = 0.0f;
  unsigned e = lane & 15u;                       // M index (output coord)
  if (e < 3u) {
    if (lane < 16u) { a.x = T[0 * 4 + e];  a.y = T[1 * 4 + e]; }   // rot[0,e], rot[1,e]
    else            { a.x = T[2 * 4 + e];  a.y = T[e * 4 + 3]; }   // rot[2,e], trans[e]
  }
  return a;
}

// Validity mask for 32 consecutive rows starting at rowBase (bit j <-> row rowBase+j).
__device__ __forceinline__ unsigned iter_mask(const float* tile, unsigned rowBase,
                                              unsigned lane, v2f A,
                                              const float* __restrict__ T) {
#if HAVE_WMMA
  (void)T;
  const unsigned sub = lane & 15u;
  const bool hi = lane >= 16u;
  unsigned m = 0;
#pragma unroll
  for (int h = 0; h < 2; ++h) {
    const unsigned r = rowBase + (unsigned)h * 16u + sub;
    // rows are 24B and 8B-aligned: three ds_load_b64, no divergent loads
    const float2* row2 = (const float2*)(tile + r * 6u);
    const float2 c01 = row2[0];                  // x, y
    const float2 c23 = row2[1];                  // z, nx
    const float2 c45 = row2[2];                  // ny, nz
    // B (4x16): VGPR0 = {x_n | z_n}, VGPR1 = {y_n | 1.0} -- branchless selects
    v2f B;
    B.x = hi ? c23.x : c01.x;
    B.y = hi ? 1.0f  : c01.y;
    v8f C = {0.f, 0.f, 0.f, 0.f, 0.f, 0.f, 0.f, 0.f};
    v8f D = __builtin_amdgcn_wmma_f32_16x16x4_f32(false, A, false, B,
                                                  (short)0, C, false, false);
    const float px = D[0], py = D[1], pz = D[2];     // lanes 0-15: point r
    const float ns = c23.y + c45.x + c45.y;
    const bool valid = (!hi) && (px * px + py * py < 1.0f) && (pz < 1.0f) && (ns != 0.0f);
    m |= (ballot32(valid) & 0xFFFFu) << (16 * h);
  }
  return m;
#else
  (void)A;
  const unsigned r = rowBase + lane;
  const float* row = tile + r * 6u;
  const float x = row[0], y = row[1], z = row[2];
  const float px = x * T[0] + y * T[4] + z * T[8]  + T[3];
  const float py = x * T[1] + y * T[5] + z * T[9]  + T[7];
  const float pz = x * T[2] + y * T[6] + z * T[10] + T[11];
  const float ns = row[3] + row[4] + row[5];
  const bool valid = (px * px + py * py < 1.0f) && (pz < 1.0f) && (ns != 0.0f);
  return ballot32(valid);
#endif
}

// ---------------- pass 1: per-block valid counts ----------------
__global__ __launch_bounds__(256) void k_count(const float* __restrict__ pc,
                                               const float* __restrict__ tt,
                                               unsigned* __restrict__ ws) {
  __shared__ float tile[TILE_F];
  __shared__ unsigned wsum[8];
  const unsigned b = blockIdx.y, blk = blockIdx.x;
  stage_tile(pc + ((size_t)b * NPTS + (size_t)blk * ROWS_PB) * 6u, tile);
  const unsigned lane = threadIdx.x & 31u, w = threadIdx.x >> 5;
  const float* T = tt + b * 16u;
  const v2f A = build_A(T, lane);
  unsigned tot = 0;
#pragma unroll
  for (int i = 0; i < 8; ++i)
    tot += __builtin_popcount(iter_mask(tile, w * 256u + (unsigned)i * 32u, lane, A, T));
  if (lane == 0) wsum[w] = tot;
  __syncthreads();
  if (threadIdx.x == 0) {
    unsigned s = 0;
#pragma unroll
    for (int i = 0; i < 8; ++i) s += wsum[i];
    ws[b * BLKS_PB + blk] = s;
  }
}

// ---------------- pass 2: per-batch exclusive scan of 64 block counts ----------------
__global__ __launch_bounds__(64) void k_scan(unsigned* __restrict__ ws) {
  __shared__ unsigned c[BLKS_PB];
  const unsigned b = blockIdx.x;
  c[threadIdx.x] = ws[b * BLKS_PB + threadIdx.x];
  __syncthreads();
  if (threadIdx.x == 0) {
    unsigned run = 0;
    for (int i = 0; i < BLKS_PB; ++i) { unsigned v = c[i]; c[i] = run; run += v; }
    ws[WS_TOTALS + b] = run;
  }
  __syncthreads();
  ws[WS_OFFSETS + b * BLKS_PB + threadIdx.x] = c[threadIdx.x];
}

// ---------------- pass 3: zero the tail rows (row >= batch total) ----------------
__global__ __launch_bounds__(256) void k_zero_tail(const unsigned* __restrict__ ws,
                                                   float* __restrict__ out) {
  const unsigned b = blockIdx.y;
  const unsigned row = blockIdx.x * 256u + threadIdx.x;
  const unsigned total = ws[WS_TOTALS + b];
  if (row < NPTS && row >= total) {
    float* dst = out + ((size_t)b * NPTS + row) * 6u;
    const float2 z2 = make_float2(0.f, 0.f);
    *(float2*)(dst + 0) = z2;
    *(float2*)(dst + 2) = z2;
    *(float2*)(dst + 4) = z2;
  }
}

// ---------------- pass 4: stable scatter of valid rows ----------------
__global__ __launch_bounds__(256) void k_scatter(const float* __restrict__ pc,
                                                 const float* __restrict__ tt,
                                                 const unsigned* __restrict__ ws,
                                                 float* __restrict__ out) {
  __shared__ float tile[TILE_F];
  __shared__ unsigned wsum[8];
  const unsigned b = blockIdx.y, blk = blockIdx.x;
  stage_tile(pc + ((size_t)b * NPTS + (size_t)blk * ROWS_PB) * 6u, tile);
  const unsigned lane = threadIdx.x & 31u, w = threadIdx.x >> 5;
  const float* T = tt + b * 16u;
  const v2f A = build_A(T, lane);
  unsigned m[8];
  unsigned tot = 0;
#pragma unroll
  for (int i = 0; i < 8; ++i) {
    m[i] = iter_mask(tile, w * 256u + (unsigned)i * 32u, lane, A, T);
    tot += __builtin_popcount(m[i]);
  }
  if (lane == 0) wsum[w] = tot;
  __syncthreads();
  unsigned run = ws[WS_OFFSETS + b * BLKS_PB + blk];   // block base within batch
  for (unsigned ww = 0; ww < w; ++ww) run += wsum[ww]; // wave prefix
  const size_t outRowBase = (size_t)b * NPTS;
#pragma unroll
  for (int i = 0; i < 8; ++i) {
    const unsigned mask = m[i];
    if ((mask >> lane) & 1u) {
      const unsigned rank = run + __builtin_popcount(mask & ((1u << lane) - 1u));
      const float2* src = (const float2*)(tile + (w * 256u + (unsigned)i * 32u + lane) * 6u);
      float* dst = out + (outRowBase + rank) * 6u;     // 24B rows, 8B aligned
      const float2 a0 = src[0], a1 = src[1], a2 = src[2];
      *(float2*)(dst + 0) = a0;
      *(float2*)(dst + 2) = a1;
      *(float2*)(dst + 4) = a2;
    }
    run += __builtin_popcount(mask);
  }
}

extern "C" void kernel_launch(void* const* d_in, const int* in_sizes, int n_in,
                              void* d_out, int out_size, void* d_ws, size_t ws_size,
                              hipStream_t stream) {
  (void)in_sizes; (void)n_in; (void)out_size; (void)ws_size;
  const float* pc = (const float*)d_in[0];   // (32, 131072, 6) f32
  const float* tt = (const float*)d_in[1];   // (32, 4, 4) f32
  float* out = (float*)d_out;                // (32, 131072, 6) f32
  unsigned* ws = (unsigned*)d_ws;

  dim3 gridC(BLKS_PB, BATCH);
  k_count  <<<gridC, 256, 0, stream>>>(pc, tt, ws);
  k_scan   <<<BATCH, 64, 0, stream>>>(ws);
  dim3 gridZ(NPTS / 256, BATCH);
  k_zero_tail<<<gridZ, 256, 0, stream>>>(ws, out);
  k_scatter<<<gridC, 256, 0, stream>>>(pc, tt, ws, out);
}